// GCN_13718125543729
// MI455X (gfx1250) — compile-verified
//
#include <hip/hip_runtime.h>

#define D 128

typedef __attribute__((ext_vector_type(2))) float v2f;
typedef __attribute__((ext_vector_type(8))) float v8f;

// ---------------------------------------------------------------------------
// Kernel 1: in-degree via native float atomic add (counts exact in f32 < 2^24)
// unsafeAtomicAdd forces GLOBAL_ATOMIC_ADD_F32 (no CAS loop).
// ---------------------------------------------------------------------------
__global__ void gcn_degree(const int* __restrict__ dst,
                           float* __restrict__ deg, int n_edges) {
  int e = blockIdx.x * blockDim.x + threadIdx.x;
  if (e < n_edges) unsafeAtomicAdd(&deg[dst[e]], 1.0f);
}

// ---------------------------------------------------------------------------
// Kernel 2: msg = feature[src] * rsqrt(deg[src]+1); summed[dst] += msg
// blockDim = (128 channels, 2 edges). Lanes of a wave cover 32 consecutive
// channels of one edge -> coalesced gather + coalesced atomic packets.
// Both feature (51.2MB) and summed (51.2MB) are L2-resident (192MB L2), so
// the ~205M non-returning f32 atomics resolve at L2 atomic rate.
// ---------------------------------------------------------------------------
__global__ void gcn_scatter(const float* __restrict__ feat,
                            const float* __restrict__ deg,
                            const int* __restrict__ src,
                            const int* __restrict__ dst,
                            float* __restrict__ summed, int n_edges) {
  int e = blockIdx.x * blockDim.y + threadIdx.y;
  if (e >= n_edges) return;
  int c = threadIdx.x;                 // 0..127
  int s = src[e];
  int d = dst[e];
  float sc = rsqrtf(deg[s] + 1.0f);
  float v = feat[(size_t)s * D + c] * sc;
  unsafeAtomicAdd(&summed[(size_t)d * D + c], v);
}

// ---------------------------------------------------------------------------
// Kernel 3: h = summed / max(deg,1)  (deg==0 rows have summed==0 -> h==0),
// then out = h @ W + b via V_WMMA_F32_16X16X4_F32 (full f32 precision; GEMM
// is only 3.3 GFLOP so the low-throughput f32 matrix path is free).
//
// One block (256 thr = 8 waves) per 16-row strip. The normalized 16x128 h
// tile is staged in LDS (row stride 132 floats -> bank-conflict-free).
// Wave w computes the 16x16 tile at columns [16w, 16w+16).
//
// ISA VGPR layouts (CDNA5 ISA):
//   A 16x4 f32 : lanes 0-15 -> M=lane, v[0]=K0,v[1]=K1 ; lanes 16-31 -> K2,K3
//   B 4x16 f32 : lanes 0-15 -> N=lane, v[0]=K0,v[1]=K1 ; lanes 16-31 -> K2,K3
//   C/D 16x16  : VGPR r, lane l -> M = r + 8*(l>=16), N = l%16
// ---------------------------------------------------------------------------
#define LDA 132

__global__ void gcn_gemm(const float* __restrict__ summed,
                         const float* __restrict__ deg,
                         const float* __restrict__ W,
                         const float* __restrict__ bias,
                         float* __restrict__ out, int n_nodes) {
  __shared__ float sA[16 * LDA];       // 8.25 KB
  const int rowBase = blockIdx.x * 16;
  const int tid = threadIdx.x;         // 0..255

  // Cooperative load + normalize: 16 rows x 128 cols = 2048 elems, 8/thread.
  #pragma unroll
  for (int i = 0; i < 8; ++i) {
    int idx = tid + i * 256;           // 0..2047
    int r = idx >> 7;                  // row within strip
    int k = idx & (D - 1);             // k index
    int n = rowBase + r;
    int nc = (n < n_nodes) ? n : (n_nodes - 1);   // clamp (no-op: 100000%16==0)
    float dg = deg[nc];
    float inv = 1.0f / fmaxf(dg, 1.0f);
    sA[r * LDA + k] = summed[(size_t)nc * D + k] * inv;
  }
  __syncthreads();

  const int wave = tid >> 5;           // 0..7 -> column tile
  const int lane = tid & 31;
  const int colBase = wave * 16;
  const int m  = lane & 15;            // M (for A) / N (for B,C,D)
  const int kh = lane >> 4;            // 0: K{0,1}, 1: K{2,3}

  v8f acc = {};
  #pragma unroll
  for (int k0 = 0; k0 < D; k0 += 4) {
    int ka = k0 + 2 * kh;
    v2f a, bm;
    a.x  = sA[m * LDA + ka];
    a.y  = sA[m * LDA + ka + 1];
    bm.x = W[(size_t)ka * D + colBase + m];
    bm.y = W[(size_t)(ka + 1) * D + colBase + m];
    // 8 args: (neg_a, A, neg_b, B, c_mod, C, reuse_a, reuse_b)
    acc = __builtin_amdgcn_wmma_f32_16x16x4_f32(
        false, a, false, bm, (short)0, acc, false, false);
  }

  float bv = bias[colBase + m];
  #pragma unroll
  for (int r = 0; r < 8; ++r) {
    int row = rowBase + r + 8 * kh;
    if (row < n_nodes)
      out[(size_t)row * D + colBase + m] = acc[r] + bv;
  }
}

// ---------------------------------------------------------------------------
// Host launch
// Inputs: 0=feature(f32 N x128), 1=W(f32 128x128), 2=b(f32 128),
//         3=src(i32 E), 4=dst(i32 E)
// Workspace: [deg: N floats][pad to 512B][summed: N*128 floats]  (~51.6 MB)
// ---------------------------------------------------------------------------
extern "C" void kernel_launch(void* const* d_in, const int* in_sizes, int n_in,
                              void* d_out, int out_size, void* d_ws, size_t ws_size,
                              hipStream_t stream) {
  const float* feature = (const float*)d_in[0];
  const float* W       = (const float*)d_in[1];
  const float* b       = (const float*)d_in[2];
  const int*   src     = (const int*)d_in[3];
  const int*   dst     = (const int*)d_in[4];
  float* out = (float*)d_out;

  const int n_nodes = in_sizes[0] / D;
  const int n_edges = in_sizes[3];

  float* deg = (float*)d_ws;
  size_t degBytes = ((size_t)n_nodes * sizeof(float) + 511) & ~(size_t)511;
  float* summed = (float*)((char*)d_ws + degBytes);
  size_t zeroBytes = degBytes + (size_t)n_nodes * D * sizeof(float);

  hipMemsetAsync(d_ws, 0, zeroBytes, stream);

  gcn_degree<<<(n_edges + 255) / 256, 256, 0, stream>>>(dst, deg, n_edges);

  dim3 sblk(D, 2);  // 256 threads = 2 edges
  gcn_scatter<<<(n_edges + 1) / 2, sblk, 0, stream>>>(feature, deg, src, dst,
                                                      summed, n_edges);

  int nStrips = (n_nodes + 15) / 16;   // 6250 for N=100000
  gcn_gemm<<<nStrips, 256, 0, stream>>>(summed, deg, W, b, out, n_nodes);
}